// LstmCrf_52536039964998
// MI455X (gfx1250) — compile-verified
//
#include <hip/hip_runtime.h>
#include <stdint.h>

typedef float        v2f __attribute__((ext_vector_type(2)));
typedef float        v8f __attribute__((ext_vector_type(8)));
typedef unsigned int v4u __attribute__((ext_vector_type(4)));
typedef int          v4i __attribute__((ext_vector_type(4)));
typedef int          v8i __attribute__((ext_vector_type(8)));

#define S_LEN   512
#define C_DIM   9
#define TS      8                 // timesteps staged per tile
#define NTILES  (S_LEN / TS)      // 64
#define WAVES_PB 4                // waves per block
#define BPW     16                // batches per wave (N dim of WMMA)
#define NEGV    (-10000.0f)
#define PADNEG  (-1.0e30f)
#define TINY    (1.0e-37f)

#define EM_ELEMS (BPW * TS * C_DIM)           // 1152 floats per emission tile
#define MK_ELEMS (BPW * TS)                   // 128 ints per mask/label tile
#define WAVE_FLOATS (2*EM_ELEMS + 4*MK_ELEMS) // double buffered: 2816 floats

// ---- fast transcendentals (raw v_exp_f32 / v_log_f32) ----------------------
#define LOG2E 1.4426950408889634f
#define LN2   0.6931471805599453f
__device__ inline float fexp(float x) { return __builtin_amdgcn_exp2f(x * LOG2E); }
__device__ inline float flog(float x) { return __builtin_amdgcn_logf(x) * LN2; }

// ---- half-wave swap (lane ^ 16) as pure VALU: v_permlanex16_b32 ------------
__device__ inline float xor16(float x) {
  int i = __float_as_int(x);
  return __int_as_float(__builtin_amdgcn_permlanex16(
      i, i, 0x76543210, (int)0xFEDCBA98, false, false));
}

// ---------------------------------------------------------------------------
// Tensor Data Mover: 2D tile (tile_d1 rows of tile_d0 elements, 4B elements,
// row stride stride0 elements) from global -> LDS, contiguous in LDS.
// ---------------------------------------------------------------------------
__device__ inline void tdm_load_2d(uint32_t lds_addr, const void* gptr,
                                   uint32_t tile_d0, uint32_t tile_d1,
                                   uint64_t stride0) {
  uint64_t ga = (uint64_t)(uintptr_t)gptr;
  v4u g0;
  g0.x = 1u;                                                   // count=1 (valid), user mode
  g0.y = lds_addr;                                             // LDS byte address
  g0.z = (uint32_t)ga;                                         // global_addr[31:0]
  g0.w = ((uint32_t)(ga >> 32) & 0x01FFFFFFu) | 0x80000000u;   // addr[56:32] | type=2
  const uint32_t td0 = 0x7FFFFFFFu, td1 = 0x7FFFFFFFu;         // huge tensor dims: no OOB clip
  v8i g1;
  g1[0] = (int)(2u << 16);                                     // workgroup_mask=0, data_size=4B
  g1[1] = (int)((td0 & 0xFFFFu) << 16);                        // tensor_dim0[15:0]
  g1[2] = (int)((td0 >> 16) | ((td1 & 0xFFFFu) << 16));        // tensor_dim0[31:16] | dim1[15:0]
  g1[3] = (int)((td1 >> 16) | (tile_d0 << 16));                // tensor_dim1[31:16] | tile_dim0
  g1[4] = (int)(tile_d1 & 0xFFFFu);                            // tile_dim1 | tile_dim2=0
  g1[5] = (int)(uint32_t)(stride0 & 0xFFFFFFFFull);            // tensor_dim0_stride[31:0]
  g1[6] = (int)(uint32_t)((stride0 >> 32) & 0xFFFFull);        // stride[47:32]
  g1[7] = 0;
  v4i z4 = {0, 0, 0, 0};
  v8i z8 = {0, 0, 0, 0, 0, 0, 0, 0};
  __builtin_amdgcn_tensor_load_to_lds(g0, g1, z4, z4, z8, 0);
}

// ---------------------------------------------------------------------------
// Forward (log-partition) + gold-path score. One wave handles 16 batches.
// State layout follows the 16x16 f32 C/D WMMA layout:
//   lane l (<16): rows(states) 0..7 for batch l ; lane l+16: rows 8..15.
// Per step: alpha' = em + m + log( expT^T x exp(alpha-m) ), done with three
// chained v_wmma_f32_16x16x4_f32 (K = states padded 9->12).
// ---------------------------------------------------------------------------
__global__ __launch_bounds__(WAVES_PB * 32) void crf_loss_fwd(
    const float* __restrict__ emission, const float* __restrict__ Tmat,
    const int* __restrict__ labels, const int* __restrict__ mask,
    float* __restrict__ diff)
{
  __shared__ float smem[WAVES_PB * WAVE_FLOATS];
  __shared__ float Ts[C_DIM * C_DIM];

  const int tid = threadIdx.x;
  if (tid < C_DIM * C_DIM) Ts[tid] = Tmat[tid];
  __syncthreads();

  const int wave   = tid >> 5;
  const int lane   = tid & 31;
  const int half   = lane >> 4;
  const int col    = lane & 15;
  const int batch0 = (blockIdx.x * WAVES_PB + wave) * BPW;
  const int batch  = batch0 + col;

  float* wbase = smem + wave * WAVE_FLOATS;
  const uint32_t ldsb = (uint32_t)(uintptr_t)(void*)wbase;  // low 32 bits = LDS offset

  // Constant A chunks: A[m,k] = exp(T[k][m]) (zero-padded), 16x4 layout:
  // lanes 0-15 hold K = {0,1}, lanes 16-31 hold K = {2,3} of each chunk.
  v2f A0, A1, A2;
  {
    const int k0 = 2 * half;
    auto aval = [&](int m, int k) -> float {
      return (m < C_DIM && k < C_DIM) ? fexp(Ts[k * C_DIM + m]) : 0.0f;
    };
    A0.x = aval(col, k0);     A0.y = aval(col, k0 + 1);
    A1.x = aval(col, 4 + k0); A1.y = aval(col, 4 + k0 + 1);
    A2.x = aval(col, 8 + k0); A2.y = aval(col, 8 + k0 + 1);
  }

  float alpha[8];
  #pragma unroll
  for (int i = 0; i < 8; ++i) {
    const int r = i + 8 * half;
    alpha[i] = (r == 0) ? 0.0f : ((r < C_DIM) ? NEGV : PADNEG);
  }

  float gscore = 0.0f;   // gold path (identical in both halves; stored by lower)
  int   glen   = 0;
  int   gprev  = 0;      // START

  // prologue: stage tile 0 (emission / mask / labels) via TDM
  tdm_load_2d(ldsb, emission + (uint64_t)batch0 * S_LEN * C_DIM,
              TS * C_DIM, BPW, (uint64_t)S_LEN * C_DIM);
  tdm_load_2d(ldsb + 4u * (2 * EM_ELEMS),
              mask + (uint64_t)batch0 * S_LEN, TS, BPW, S_LEN);
  tdm_load_2d(ldsb + 4u * (2 * EM_ELEMS + 2 * MK_ELEMS),
              labels + (uint64_t)batch0 * S_LEN, TS, BPW, S_LEN);

  for (int g = 0; g < NTILES; ++g) {
    const int buf = g & 1;
    if (g + 1 < NTILES) {
      const int nb = buf ^ 1;
      tdm_load_2d(ldsb + 4u * (nb * EM_ELEMS),
                  emission + ((uint64_t)batch0 * S_LEN + (g + 1) * TS) * C_DIM,
                  TS * C_DIM, BPW, (uint64_t)S_LEN * C_DIM);
      tdm_load_2d(ldsb + 4u * (2 * EM_ELEMS + nb * MK_ELEMS),
                  mask + (uint64_t)batch0 * S_LEN + (g + 1) * TS, TS, BPW, S_LEN);
      tdm_load_2d(ldsb + 4u * (2 * EM_ELEMS + 2 * MK_ELEMS + nb * MK_ELEMS),
                  labels + (uint64_t)batch0 * S_LEN + (g + 1) * TS, TS, BPW, S_LEN);
      __builtin_amdgcn_s_wait_tensorcnt(3);   // current tile's 3 loads retired
    } else {
      __builtin_amdgcn_s_wait_tensorcnt(0);
    }
    asm volatile("" ::: "memory");

    const float* emT = wbase + buf * EM_ELEMS;
    const float* mkT = wbase + 2 * EM_ELEMS + buf * MK_ELEMS;
    const float* lbT = wbase + 2 * EM_ELEMS + 2 * MK_ELEMS + buf * MK_ELEMS;

    for (int toff = 0; toff < TS; ++toff) {
      const int t     = g * TS + toff;
      const int mt    = __float_as_int(mkT[col * TS + toff]);
      const int ebase = col * (TS * C_DIM) + toff * C_DIM;

      if (__builtin_expect(t == 0, 0)) {
        // exact log-space first step (alpha0 = delta at START; spread = 1e4).
        // Padded rows MUST be pinned at PADNEG here so they cannot hijack the
        // exp-space normalizer at t==1.
        #pragma unroll
        for (int i = 0; i < 8; ++i) {
          const int  r     = i + 8 * half;
          const bool valid = (half == 0) | (i == 0);
          const int  ri    = valid ? r : 0;
          float mm = -3.0e38f;
          #pragma unroll
          for (int cp = 0; cp < C_DIM; ++cp)
            mm = fmaxf(mm, ((cp == 0) ? 0.0f : NEGV) + Ts[cp * C_DIM + ri]);
          float ssum = 0.0f;
          #pragma unroll
          for (int cp = 0; cp < C_DIM; ++cp)
            ssum += fexp(((cp == 0) ? 0.0f : NEGV) + Ts[cp * C_DIM + ri] - mm);
          float na = emT[ebase + ri] + mm + flog(ssum);
          na = mt ? na : alpha[i];
          alpha[i] = valid ? na : PADNEG;
        }
      } else {
        // exp-space step with global renormalization + 3x WMMA f32 16x16x4
        float mloc = alpha[0];
        #pragma unroll
        for (int i = 1; i < 8; ++i) mloc = fmaxf(mloc, alpha[i]);
        const float m = fmaxf(mloc, xor16(mloc));

        float p[8];
        #pragma unroll
        for (int i = 0; i < 8; ++i) p[i] = fexp(alpha[i] - m);

        // B chunks, 4x16 layout: VGPR0 = rows {K, K+2} split low/high lanes.
        const float s0 = xor16(p[0]);
        const float s2 = xor16(p[2]);
        const float s3 = xor16(p[3]);
        const float s6 = xor16(p[6]);
        const float s7 = xor16(p[7]);
        v2f B0, B1, B2;
        B0.x = half ? s2 : p[0];  B0.y = half ? s3 : p[1];
        B1.x = half ? s6 : p[4];  B1.y = half ? s7 : p[5];
        B2.x = half ? 0.0f : s0;  B2.y = 0.0f;   // states 9..11 are exact zeros

        v8f acc = {0.f, 0.f, 0.f, 0.f, 0.f, 0.f, 0.f, 0.f};
        acc = __builtin_amdgcn_wmma_f32_16x16x4_f32(false, A0, false, B0, (short)0, acc, false, false);
        acc = __builtin_amdgcn_wmma_f32_16x16x4_f32(false, A1, false, B1, (short)0, acc, false, false);
        acc = __builtin_amdgcn_wmma_f32_16x16x4_f32(false, A2, false, B2, (short)0, acc, false, false);

        // Phase 1: compute all new alphas unconditionally (nothing to sink
        // under a divergent branch). Padded rows get acc==0 -> m-85+-6: inert.
        float na[8];
        #pragma unroll
        for (int i = 0; i < 8; ++i) {
          const float e = emT[ebase + i + 8 * half];
          na[i] = e + m + flog(fmaxf(acc[i], TINY));
        }
        // Phase 2: bare register selects on the mask -> v_cndmask, no branches.
        #pragma unroll
        for (int i = 0; i < 8; ++i)
          alpha[i] = mt ? na[i] : alpha[i];
      }

      // gold-path score: inputs depend only on col -> identical in both halves
      {
        const int   lbl = __float_as_int(lbT[col * TS + toff]);
        const float e   = emT[ebase + lbl];
        const float tr  = Ts[gprev * C_DIM + lbl];
        gscore += (e + tr) * (float)mt;
        glen   += mt;
        gprev   = lbl;
      }
    }
  }

  // log_Z = logsumexp_r( alpha[r] + T[r][END] ); pads masked out here.
  float fl[8];
  #pragma unroll
  for (int i = 0; i < 8; ++i) {
    const int  r     = i + 8 * half;
    const bool valid = (half == 0) | (i == 0);
    const int  ri    = valid ? r : 0;
    const float v = alpha[i] + Ts[ri * C_DIM + (C_DIM - 1)];
    fl[i] = valid ? v : PADNEG;
  }
  float mloc = fl[0];
  #pragma unroll
  for (int i = 1; i < 8; ++i) mloc = fmaxf(mloc, fl[i]);
  const float mm = fmaxf(mloc, xor16(mloc));
  float ssum = 0.0f;
  #pragma unroll
  for (int i = 0; i < 8; ++i) ssum += fexp(fl[i] - mm);
  ssum += xor16(ssum);
  const float logZ = mm + flog(ssum);

  if (half == 0) {
    const int   last = (glen == 0) ? 0 : labels[(uint64_t)batch * S_LEN + (glen - 1)];
    const float gold = gscore + Ts[last * C_DIM + (C_DIM - 1)];
    diff[batch] = logZ - gold;
  }
}

// ---------------------------------------------------------------------------
__global__ void reduce_mean(const float* __restrict__ in, float* __restrict__ out, int n) {
  __shared__ float red[256];
  float s = 0.0f;
  for (int i = threadIdx.x; i < n; i += 256) s += in[i];
  red[threadIdx.x] = s;
  __syncthreads();
  for (int k = 128; k > 0; k >>= 1) {
    if ((int)threadIdx.x < k) red[threadIdx.x] += red[threadIdx.x + k];
    __syncthreads();
  }
  if (threadIdx.x == 0) out[0] = red[0] / (float)n;
}

extern "C" void kernel_launch(void* const* d_in, const int* in_sizes, int n_in,
                              void* d_out, int out_size, void* d_ws, size_t ws_size,
                              hipStream_t stream) {
  const float* emission = (const float*)d_in[0];
  const float* T        = (const float*)d_in[1];
  const int*   labels   = (const int*)d_in[2];
  const int*   mask     = (const int*)d_in[3];
  float* diff = (float*)d_ws;                  // B floats of (logZ - gold)
  float* out  = (float*)d_out;

  const int B = in_sizes[2] / S_LEN;           // labels is B*S
  const int blocks = B / (WAVES_PB * BPW);     // 8192 / 64 = 128

  crf_loss_fwd<<<blocks, WAVES_PB * 32, 0, stream>>>(emission, T, labels, mask, diff);
  reduce_mean<<<1, 256, 0, stream>>>(diff, out, B);
}